// InductiveSAGEEncoder_62311385530525
// MI455X (gfx1250) — compile-verified
//
#include <hip/hip_runtime.h>
#include <cstdint>

typedef __attribute__((ext_vector_type(2))) float v2f;
typedef __attribute__((ext_vector_type(8))) float v8f;

#define N1_NODES 90112   // num_dst1
#define N2_NODES 8192    // num_dst2
#define DIN  128
#define DH   256
#define DOUT 128

// ---------------------------------------------------------------------------
// Edge scatter-add: agg[dst[e]][f] += feat[src[e]][f]; cnt[dst[e]] += 1.
// Thread layout: consecutive threads = consecutive features of one edge, so
// both the gather read and the atomic write are fully coalesced rows.
// ---------------------------------------------------------------------------
__global__ void sage_scatter_add(const float* __restrict__ feat,
                                 const int*   __restrict__ src,
                                 const int*   __restrict__ dst,
                                 float* __restrict__ agg,
                                 float* __restrict__ cnt,
                                 long long total, int D) {
    long long gid = (long long)blockIdx.x * blockDim.x + threadIdx.x;
    if (gid >= total) return;
    int e = (int)(gid / D);
    int f = (int)(gid - (long long)e * D);
    int s = src[e];
    int d = dst[e];
    atomicAdd(&agg[(long long)d * D + f], feat[(long long)s * D + f]);
    if (f == 0) atomicAdd(&cnt[d], 1.0f);
}

// agg[i][f] /= max(cnt[i], 1)
__global__ void sage_seg_div(float* __restrict__ agg,
                             const float* __restrict__ cnt,
                             long long total, int D) {
    long long gid = (long long)blockIdx.x * blockDim.x + threadIdx.x;
    if (gid >= total) return;
    int i = (int)(gid / D);
    agg[gid] = agg[gid] * (1.0f / fmaxf(cnt[i], 1.0f));
}

// ---------------------------------------------------------------------------
// Fused dual-GEMM epilogue:  C = act(A1 @ B1 + A2 @ B2 + bias)
//   A1,A2: [M,K] row-major   B1,B2: [K,N] row-major   C: [M,N]
// One wave owns a 16 x (16*NSUB) output strip; both products accumulate into
// the same v_wmma_f32_16x16x4_f32 accumulators (A fragments reused NSUB x).
// ---------------------------------------------------------------------------
template <int NSUB>
__global__ void sage_gemm_wmma(const float* __restrict__ A1,
                               const float* __restrict__ B1,
                               const float* __restrict__ A2,
                               const float* __restrict__ B2,
                               const float* __restrict__ bias,
                               float* __restrict__ C,
                               int M, int N, int K, int do_relu) {
    const int lane = threadIdx.x & 31;
    const int wave = threadIdx.x >> 5;
    const int wavesPerBlock = blockDim.x >> 5;

    const int tilesN = N / (16 * NSUB);
    const int w = blockIdx.x * wavesPerBlock + wave;
    if (w >= (M / 16) * tilesN) return;          // wave-uniform: EXEC stays all-1s

    const int tm   = w / tilesN;
    const int tn   = w - tm * tilesN;
    const int row0 = tm * 16;
    const int col0 = tn * 16 * NSUB;
    const int half = lane >> 4;                  // K-pair selector (ISA A/B layout)
    const int lm   = lane & 15;

    v8f acc[NSUB] = {};

    // A fragment: lane holds A[row0+lm, k0 + 2*half + {0,1}] -> aligned float2
    const float* a1p = A1 + (long long)(row0 + lm) * K + 2 * half;
    const float* a2p = A2 + (long long)(row0 + lm) * K + 2 * half;

    for (int k0 = 0; k0 < K; k0 += 4) {
        v2f a1 = *(const v2f*)(a1p + k0);
        v2f a2 = *(const v2f*)(a2p + k0);
        const int kb = k0 + 2 * half;
#pragma unroll
        for (int s = 0; s < NSUB; ++s) {
            const int col = col0 + s * 16 + lm;
            v2f b1, b2;
            b1.x = B1[(long long)kb * N + col];
            b1.y = B1[(long long)(kb + 1) * N + col];
            b2.x = B2[(long long)kb * N + col];
            b2.y = B2[(long long)(kb + 1) * N + col];
            // (neg_a, A, neg_b, B, c_mod, C, reuse_a, reuse_b)
            acc[s] = __builtin_amdgcn_wmma_f32_16x16x4_f32(
                false, a1, false, b1, (short)0, acc[s], false, false);
            acc[s] = __builtin_amdgcn_wmma_f32_16x16x4_f32(
                false, a2, false, b2, (short)0, acc[s], false, false);
        }
    }

#pragma unroll
    for (int s = 0; s < NSUB; ++s) {
        const int col = col0 + s * 16 + lm;
        const float b = bias[col];
#pragma unroll
        for (int r = 0; r < 8; ++r) {
            const int row = row0 + r + 8 * half;  // C/D VGPR layout
            float v = acc[s][r] + b;
            if (do_relu) v = fmaxf(v, 0.0f);
            C[(long long)row * N + col] = v;
        }
    }
}

// ---------------------------------------------------------------------------
extern "C" void kernel_launch(void* const* d_in, const int* in_sizes, int n_in,
                              void* d_out, int out_size, void* d_ws, size_t ws_size,
                              hipStream_t stream) {
    const float* x    = (const float*)d_in[0];   // [N_TOTAL, 128]
    const float* Wl1  = (const float*)d_in[1];   // [128, 256]
    const float* bl1  = (const float*)d_in[2];   // [256]
    const float* Wr1  = (const float*)d_in[3];   // [128, 256]
    const float* Wl2  = (const float*)d_in[4];   // [256, 128]
    const float* bl2  = (const float*)d_in[5];   // [128]
    const float* Wr2  = (const float*)d_in[6];   // [256, 128]
    const int*   src1 = (const int*)d_in[7];
    const int*   dst1 = (const int*)d_in[8];
    const int*   src2 = (const int*)d_in[9];
    const int*   dst2 = (const int*)d_in[10];
    const long long E1 = in_sizes[7];
    const long long E2 = in_sizes[9];

    // Workspace layout (floats):
    float* agg1 = (float*)d_ws;                          // [N1, 128]
    float* cnt1 = agg1 + (size_t)N1_NODES * DIN;         // [N1]
    float* h    = cnt1 + N1_NODES;                       // [N1, 256]
    float* agg2 = h    + (size_t)N1_NODES * DH;          // [N2, 256]
    float* cnt2 = agg2 + (size_t)N2_NODES * DH;          // [N2]

    // Zero accumulators (agg+cnt are contiguous per layer).
    hipMemsetAsync(agg1, 0, sizeof(float) * ((size_t)N1_NODES * DIN + N1_NODES), stream);
    hipMemsetAsync(agg2, 0, sizeof(float) * ((size_t)N2_NODES * DH  + N2_NODES), stream);

    // ---- Layer 1: mean-aggregate over E1 edges into agg1 ----
    {
        long long total = E1 * DIN;
        int blocks = (int)((total + 255) / 256);
        sage_scatter_add<<<blocks, 256, 0, stream>>>(x, src1, dst1, agg1, cnt1, total, DIN);
    }
    {
        long long total = (long long)N1_NODES * DIN;
        int blocks = (int)((total + 255) / 256);
        sage_seg_div<<<blocks, 256, 0, stream>>>(agg1, cnt1, total, DIN);
    }
    // h = relu(agg1 @ Wl1 + x[:N1] @ Wr1 + bl1)   [N1, 256]
    {
        int tiles  = (N1_NODES / 16) * (DH / (16 * 4));   // 16x64 strip per wave
        int blocks = (tiles + 7) / 8;                     // 8 waves / 256-thread block
        sage_gemm_wmma<4><<<blocks, 256, 0, stream>>>(
            agg1, Wl1, x, Wr1, bl1, h, N1_NODES, DH, DIN, /*relu=*/1);
    }

    // ---- Layer 2: mean-aggregate over E2 edges into agg2 ----
    {
        long long total = E2 * DH;
        int blocks = (int)((total + 255) / 256);
        sage_scatter_add<<<blocks, 256, 0, stream>>>(h, src2, dst2, agg2, cnt2, total, DH);
    }
    {
        long long total = (long long)N2_NODES * DH;
        int blocks = (int)((total + 255) / 256);
        sage_seg_div<<<blocks, 256, 0, stream>>>(agg2, cnt2, total, DH);
    }
    // out = agg2 @ Wl2 + h[:N2] @ Wr2 + bl2   [N2, 128]
    {
        int tiles  = (N2_NODES / 16) * (DOUT / (16 * 2)); // 16x32 strip per wave
        int blocks = (tiles + 7) / 8;
        sage_gemm_wmma<2><<<blocks, 256, 0, stream>>>(
            agg2, Wl2, h, Wr2, bl2, (float*)d_out, N2_NODES, DOUT, DH, /*relu=*/0);
    }
}